// Net_55533927137975
// MI455X (gfx1250) — compile-verified
//
#include <hip/hip_runtime.h>
#include <hip/hip_bf16.h>
#include <math.h>

// ---------------------------------------------------------------------------
// 2-layer GAT for MI455X (gfx1250).  GEMMs via V_WMMA_F32_16X16X4_F32 (exact
// fp32) with one wave owning a full output stripe (A fragment loaded once per
// K-step, 4 independent WMMAs back-to-back).  Edge softmax/aggregation via
// L2-resident float atomics (all per-node state fits in the 192MB L2).
// ---------------------------------------------------------------------------

#define F_IN   512
#define H1     8
#define D1     8
#define HD1    64     // H1*D1
#define C_OUT  40
#define NEG_SLOPE 0.2f

typedef float v2f __attribute__((ext_vector_type(2)));
typedef float v8f __attribute__((ext_vector_type(8)));

#define WMMA_F32(a, b, c) __builtin_amdgcn_wmma_f32_16x16x4_f32( \
        false, (a), false, (b), (short)0, (c), false, false)

// ----- monotone float<->uint key mapping for atomic max on signed floats ----
__device__ __forceinline__ unsigned fkey(float f) {
    unsigned b = __float_as_uint(f);
    return (b & 0x80000000u) ? ~b : (b | 0x80000000u);
}
__device__ __forceinline__ float funkey(unsigned k) {
    unsigned b = (k & 0x80000000u) ? (k & 0x7fffffffu) : ~k;
    return __uint_as_float(b);
}
#define FKEY_NEGINF 0x007FFFFFu   // fkey(-inf)

__device__ __forceinline__ float leaky(float e) {
    return e >= 0.0f ? e : NEG_SLOPE * e;
}

// ===========================================================================
// GEMM1: h1[N,64] = x[N,512] @ W1[512,64]
// Block = 128 threads = 4 waves; wave w owns the 16-row stripe
// rows [64*blockIdx.x + 16*w, +16) x all 64 columns (4 accumulators).
// A 16x4 frag:  lane m=l&15 holds row m, K = k + (l>=16 ? 2:0) .. +1
// B  4x16 frag: lane n=l&15 holds col n, same K split.
// ===========================================================================
__global__ __launch_bounds__(128) void gemm1_wmma(
    const float* __restrict__ x, const float* __restrict__ W1,
    float* __restrict__ h1, int N)
{
    const int wave  = threadIdx.x >> 5;
    const int lane  = threadIdx.x & 31;
    const int row0  = (blockIdx.x * 4 + wave) * 16;
    const int mn    = lane & 15;
    const int khalf = (lane >> 4) << 1;     // 0 or 2
    if (row0 >= N) return;

    int arow = row0 + mn; if (arow >= N) arow = N - 1;   // clamp OOB reads
    const float* __restrict__ ap = x + (size_t)arow * F_IN;

    v8f c0 = {}, c1 = {}, c2 = {}, c3 = {};
    #pragma unroll 4
    for (int k = 0; k < F_IN; k += 4) {
        v2f a;
        a.x = ap[k + khalf];
        a.y = ap[k + khalf + 1];
        const float* bp = W1 + (size_t)(k + khalf) * HD1 + mn;
        v2f b0, b1, b2, b3;
        b0.x = bp[0];       b0.y = bp[HD1];
        b1.x = bp[16];      b1.y = bp[HD1 + 16];
        b2.x = bp[32];      b2.y = bp[HD1 + 32];
        b3.x = bp[48];      b3.y = bp[HD1 + 48];
        c0 = WMMA_F32(a, b0, c0);
        c1 = WMMA_F32(a, b1, c1);
        c2 = WMMA_F32(a, b2, c2);
        c3 = WMMA_F32(a, b3, c3);
    }
    // C/D layout: VGPR r -> lanes0-15:(M=r,N=lane)  lanes16-31:(M=r+8,N=lane-16)
    const int mbase = (lane >> 4) << 3;
    float* hp = h1 + (size_t)(row0 + mbase) * HD1 + mn;
    if (row0 + 16 <= N) {                       // full tile: unguarded stores
        #pragma unroll
        for (int r = 0; r < 8; ++r) {
            float* p = hp + (size_t)r * HD1;
            p[0] = c0[r]; p[16] = c1[r]; p[32] = c2[r]; p[48] = c3[r];
        }
    } else {
        for (int r = 0; r < 8; ++r) {
            if (row0 + mbase + r < N) {
                float* p = hp + (size_t)r * HD1;
                p[0] = c0[r]; p[16] = c1[r]; p[32] = c2[r]; p[48] = c3[r];
            }
        }
    }
}

// ===========================================================================
// GEMM2: h2[N,40] = h1act[N,64] @ W2[64,40]
// One wave per 16-row stripe; 3 accumulators cover 48 padded cols (40 valid).
// ===========================================================================
__global__ __launch_bounds__(32) void gemm2_wmma(
    const float* __restrict__ A, const float* __restrict__ W2,
    float* __restrict__ h2, int N)
{
    const int lane  = threadIdx.x & 31;
    const int row0  = blockIdx.x * 16;
    const int mn    = lane & 15;
    const int khalf = (lane >> 4) << 1;
    if (row0 >= N) return;

    int arow = row0 + mn; if (arow >= N) arow = N - 1;
    const float* __restrict__ ap = A + (size_t)arow * HD1;
    const bool tail = (32 + mn) < C_OUT;        // last tile: cols 32..39 valid

    v8f c0 = {}, c1 = {}, c2 = {};
    #pragma unroll
    for (int k = 0; k < HD1; k += 4) {
        v2f a;
        a.x = ap[k + khalf];
        a.y = ap[k + khalf + 1];
        const float* bp = W2 + (size_t)(k + khalf) * C_OUT + mn;
        v2f b0, b1, b2;
        b0.x = bp[0];                    b0.y = bp[C_OUT];
        b1.x = bp[16];                   b1.y = bp[C_OUT + 16];
        b2.x = tail ? bp[32] : 0.f;      b2.y = tail ? bp[C_OUT + 32] : 0.f;
        c0 = WMMA_F32(a, b0, c0);
        c1 = WMMA_F32(a, b1, c1);
        c2 = WMMA_F32(a, b2, c2);
    }
    const int mbase = (lane >> 4) << 3;
    float* hp = h2 + (size_t)(row0 + mbase) * C_OUT + mn;
    if (row0 + 16 <= N) {
        #pragma unroll
        for (int r = 0; r < 8; ++r) {
            float* p = hp + (size_t)r * C_OUT;
            p[0] = c0[r]; p[16] = c1[r];
            if (tail) p[32] = c2[r];
        }
    } else {
        for (int r = 0; r < 8; ++r) {
            if (row0 + mbase + r < N) {
                float* p = hp + (size_t)r * C_OUT;
                p[0] = c0[r]; p[16] = c1[r];
                if (tail) p[32] = c2[r];
            }
        }
    }
}

// ===========================================================================
// Layer-1 attention logits + state init.  One thread per (node, head).
// ===========================================================================
__global__ void att1_init(
    const float* __restrict__ h1,
    const float* __restrict__ a_src, const float* __restrict__ a_dst,
    float* __restrict__ als, float* __restrict__ ald,
    unsigned* __restrict__ m1, float* __restrict__ s1,
    float* __restrict__ o1, int N)
{
    int idx = blockIdx.x * blockDim.x + threadIdx.x;
    if (idx >= N * H1) return;
    int n = idx >> 3, h = idx & 7;
    const float* hv = h1 + (size_t)n * HD1 + h * D1;
    float as = 0.f, ad = 0.f;
    #pragma unroll
    for (int d = 0; d < D1; ++d) {
        as += hv[d] * a_src[h * D1 + d];
        ad += hv[d] * a_dst[h * D1 + d];
    }
    als[idx] = as;
    ald[idx] = ad;
    m1[idx]  = FKEY_NEGINF;
    s1[idx]  = 0.f;
    float* op = o1 + (size_t)n * HD1 + h * D1;
    #pragma unroll
    for (int d = 0; d < D1; ++d) op[d] = 0.f;
}

__device__ __forceinline__ void edge_nodes(const long long* __restrict__ ei,
                                           int t, int E, int& src, int& dst) {
    if (t < E) { src = (int)ei[t]; dst = (int)ei[(size_t)E + t]; }
    else       { src = dst = t - E; }               // self-loop
}

// ----- pass 1: segment max ------------------------------------------------
__global__ void edge1_max(const long long* __restrict__ ei, int E, int N,
                          const float* __restrict__ als,
                          const float* __restrict__ ald,
                          unsigned* __restrict__ m1)
{
    int t = blockIdx.x * blockDim.x + threadIdx.x;
    if (t >= E + N) return;
    int src, dst; edge_nodes(ei, t, E, src, dst);
    #pragma unroll
    for (int h = 0; h < H1; ++h) {
        float e = leaky(als[src * H1 + h] + ald[dst * H1 + h]);
        atomicMax(&m1[dst * H1 + h], fkey(e));
    }
}

// ----- pass 2: segment sum of exp ----------------------------------------
__global__ void edge1_sum(const long long* __restrict__ ei, int E, int N,
                          const float* __restrict__ als,
                          const float* __restrict__ ald,
                          const unsigned* __restrict__ m1,
                          float* __restrict__ s1)
{
    int t = blockIdx.x * blockDim.x + threadIdx.x;
    if (t >= E + N) return;
    int src, dst; edge_nodes(ei, t, E, src, dst);
    #pragma unroll
    for (int h = 0; h < H1; ++h) {
        float e = leaky(als[src * H1 + h] + ald[dst * H1 + h]);
        float ex = expf(e - funkey(m1[dst * H1 + h]));
        atomicAdd(&s1[dst * H1 + h], ex);
    }
}

// ----- pass 3: alpha-weighted scatter of messages; thread per (edge,head) --
__global__ void edge1_msg(const long long* __restrict__ ei, int E, int N,
                          const float* __restrict__ als,
                          const float* __restrict__ ald,
                          const unsigned* __restrict__ m1,
                          const float* __restrict__ s1,
                          const float* __restrict__ h1,
                          float* __restrict__ o1)
{
    int idx = blockIdx.x * blockDim.x + threadIdx.x;
    if (idx >= (E + N) * H1) return;
    int t = idx >> 3, h = idx & 7;
    int src, dst; edge_nodes(ei, t, E, src, dst);
    float e = leaky(als[src * H1 + h] + ald[dst * H1 + h]);
    float alpha = expf(e - funkey(m1[dst * H1 + h])) / (s1[dst * H1 + h] + 1e-16f);
    const float4* hv = (const float4*)(h1 + (size_t)src * HD1 + h * D1);
    float4 v0 = hv[0], v1 = hv[1];
    float* op = o1 + (size_t)dst * HD1 + h * D1;
    atomicAdd(op + 0, alpha * v0.x); atomicAdd(op + 1, alpha * v0.y);
    atomicAdd(op + 2, alpha * v0.z); atomicAdd(op + 3, alpha * v0.w);
    atomicAdd(op + 4, alpha * v1.x); atomicAdd(op + 5, alpha * v1.y);
    atomicAdd(op + 6, alpha * v1.z); atomicAdd(op + 7, alpha * v1.w);
}

// ----- bias + ELU (in place: o1 becomes h1act) ----------------------------
__global__ void bias_elu(float* __restrict__ o1, const float* __restrict__ b1, int N)
{
    int idx = blockIdx.x * blockDim.x + threadIdx.x;
    if (idx >= N * HD1) return;
    float v = o1[idx] + b1[idx & (HD1 - 1)];
    o1[idx] = v > 0.f ? v : expf(v) - 1.f;
}

// ----- layer-2 attention logits + state init (one thread per node) --------
__global__ void att2_init(const float* __restrict__ h2,
                          const float* __restrict__ a_src,
                          const float* __restrict__ a_dst,
                          float* __restrict__ als, float* __restrict__ ald,
                          unsigned* __restrict__ m2, float* __restrict__ s2,
                          float* __restrict__ o2, int N)
{
    int n = blockIdx.x * blockDim.x + threadIdx.x;
    if (n >= N) return;
    const float* hv = h2 + (size_t)n * C_OUT;
    float as = 0.f, ad = 0.f;
    for (int d = 0; d < C_OUT; ++d) {
        as += hv[d] * a_src[d];
        ad += hv[d] * a_dst[d];
    }
    als[n] = as; ald[n] = ad;
    m2[n] = FKEY_NEGINF; s2[n] = 0.f;
    float* op = o2 + (size_t)n * C_OUT;
    for (int d = 0; d < C_OUT; ++d) op[d] = 0.f;
}

__global__ void edge2_max(const long long* __restrict__ ei, int E, int N,
                          const float* __restrict__ als,
                          const float* __restrict__ ald,
                          unsigned* __restrict__ m2)
{
    int t = blockIdx.x * blockDim.x + threadIdx.x;
    if (t >= E + N) return;
    int src, dst; edge_nodes(ei, t, E, src, dst);
    atomicMax(&m2[dst], fkey(leaky(als[src] + ald[dst])));
}

__global__ void edge2_sum(const long long* __restrict__ ei, int E, int N,
                          const float* __restrict__ als,
                          const float* __restrict__ ald,
                          const unsigned* __restrict__ m2,
                          float* __restrict__ s2)
{
    int t = blockIdx.x * blockDim.x + threadIdx.x;
    if (t >= E + N) return;
    int src, dst; edge_nodes(ei, t, E, src, dst);
    float e = leaky(als[src] + ald[dst]);
    atomicAdd(&s2[dst], expf(e - funkey(m2[dst])));
}

// thread per (edge, chunk-of-5): 8 chunks cover the 40 output dims
__global__ void edge2_msg(const long long* __restrict__ ei, int E, int N,
                          const float* __restrict__ als,
                          const float* __restrict__ ald,
                          const unsigned* __restrict__ m2,
                          const float* __restrict__ s2,
                          const float* __restrict__ h2,
                          float* __restrict__ o2)
{
    int idx = blockIdx.x * blockDim.x + threadIdx.x;
    if (idx >= (E + N) * 8) return;
    int t = idx >> 3, ch = idx & 7;
    int src, dst; edge_nodes(ei, t, E, src, dst);
    float e = leaky(als[src] + ald[dst]);
    float alpha = expf(e - funkey(m2[dst])) / (s2[dst] + 1e-16f);
    const float* hv = h2 + (size_t)src * C_OUT + ch * 5;
    float* op = o2 + (size_t)dst * C_OUT + ch * 5;
    #pragma unroll
    for (int j = 0; j < 5; ++j) atomicAdd(op + j, alpha * hv[j]);
}

// ----- bias + log_softmax over 40 classes (one thread per node) -----------
__global__ void final_logsoftmax(const float* __restrict__ o2,
                                 const float* __restrict__ b2,
                                 float* __restrict__ out, int N)
{
    int n = blockIdx.x * blockDim.x + threadIdx.x;
    if (n >= N) return;
    float v[C_OUT];
    float mx = -INFINITY;
    const float* ip = o2 + (size_t)n * C_OUT;
    for (int c = 0; c < C_OUT; ++c) { v[c] = ip[c] + b2[c]; mx = fmaxf(mx, v[c]); }
    float s = 0.f;
    for (int c = 0; c < C_OUT; ++c) s += expf(v[c] - mx);
    float lse = logf(s);
    float* op = out + (size_t)n * C_OUT;
    for (int c = 0; c < C_OUT; ++c) op[c] = v[c] - mx - lse;
}

// ===========================================================================
extern "C" void kernel_launch(void* const* d_in, const int* in_sizes, int n_in,
                              void* d_out, int out_size, void* d_ws, size_t ws_size,
                              hipStream_t stream)
{
    const float*     x      = (const float*)d_in[0];
    const long long* ei     = (const long long*)d_in[1];   // int64 per reference
    const float*     W1     = (const float*)d_in[2];
    const float*     a_src1 = (const float*)d_in[3];
    const float*     a_dst1 = (const float*)d_in[4];
    const float*     b1     = (const float*)d_in[5];
    const float*     W2     = (const float*)d_in[6];
    const float*     a_src2 = (const float*)d_in[7];
    const float*     a_dst2 = (const float*)d_in[8];
    const float*     b2     = (const float*)d_in[9];
    float*           out    = (float*)d_out;

    const int N = in_sizes[0] / F_IN;
    const int E = in_sizes[1] / 2;
    const int T = E + N;                 // edges incl. self-loops

    // -------- workspace layout (floats) ----------------------------------
    float* ws = (float*)d_ws;
    size_t o = 0;
    float*    h1   = ws + o; o += (size_t)N * HD1;
    float*    als1 = ws + o; o += (size_t)N * H1;
    float*    ald1 = ws + o; o += (size_t)N * H1;
    unsigned* m1   = (unsigned*)(ws + o); o += (size_t)N * H1;
    float*    s1   = ws + o; o += (size_t)N * H1;
    float*    o1   = ws + o; o += (size_t)N * HD1;   // -> h1act after bias_elu
    float*    h2   = ws + o; o += (size_t)N * C_OUT;
    float*    als2 = ws + o; o += (size_t)N;
    float*    ald2 = ws + o; o += (size_t)N;
    unsigned* m2   = (unsigned*)(ws + o); o += (size_t)N;
    float*    s2   = ws + o; o += (size_t)N;
    float*    o2   = ws + o; o += (size_t)N * C_OUT;
    (void)ws_size; (void)n_in; (void)out_size;

    const int rowTiles = (N + 15) / 16;
    const int B = 256;

    // -------- layer 1 -----------------------------------------------------
    gemm1_wmma<<<(N + 63) / 64, 128, 0, stream>>>(x, W1, h1, N);
    att1_init<<<(N * H1 + B - 1) / B, B, 0, stream>>>(h1, a_src1, a_dst1,
                                                      als1, ald1, m1, s1, o1, N);
    edge1_max<<<(T + B - 1) / B, B, 0, stream>>>(ei, E, N, als1, ald1, m1);
    edge1_sum<<<(T + B - 1) / B, B, 0, stream>>>(ei, E, N, als1, ald1, m1, s1);
    edge1_msg<<<((size_t)T * H1 + B - 1) / B, B, 0, stream>>>(ei, E, N, als1, ald1,
                                                              m1, s1, h1, o1);
    bias_elu<<<(N * HD1 + B - 1) / B, B, 0, stream>>>(o1, b1, N);

    // -------- layer 2 -----------------------------------------------------
    gemm2_wmma<<<rowTiles, 32, 0, stream>>>(o1, W2, h2, N);
    att2_init<<<(N + B - 1) / B, B, 0, stream>>>(h2, a_src2, a_dst2,
                                                 als2, ald2, m2, s2, o2, N);
    edge2_max<<<(T + B - 1) / B, B, 0, stream>>>(ei, E, N, als2, ald2, m2);
    edge2_sum<<<(T + B - 1) / B, B, 0, stream>>>(ei, E, N, als2, ald2, m2, s2);
    edge2_msg<<<((size_t)T * 8 + B - 1) / B, B, 0, stream>>>(ei, E, N, als2, ald2,
                                                             m2, s2, h2, o2);
    final_logsoftmax<<<(N + B - 1) / B, B, 0, stream>>>(o2, b2, out, N);
}